// NCCWithGrad_5643587026991
// MI455X (gfx1250) — compile-verified
//
#include <hip/hip_runtime.h>
#include <hip/hip_bf16.h>
#include <stdint.h>

// ---------------------------------------------------------------------------
// NCC (9x9 local normalized cross-correlation) + finite-difference grad loss.
// Memory-bound: stage tiles via gfx1250 async global->LDS (b128 vectorized),
// vertical box sum via V_WMMA_F32_16X16X4_F32 (banded-ones matmul),
// horizontal box sum via per-thread running sum. 32x128 output tile / block.
// ---------------------------------------------------------------------------

typedef __attribute__((ext_vector_type(2))) float v2f;
typedef __attribute__((ext_vector_type(8))) float v8f;

#define IMG_H 512
#define IMG_W 512
#define TILE_ROWS 32          // output rows per block
#define TILE_COLS 128         // output cols per block
#define TH 40                 // staged rows  (32 + 2*4 halo)
#define TW 144                // staged cols  (128 + 8 halo = 136, pad to 9*16)
#define NTHREADS 512          // 16 waves
#define WIN_AREA 81.0f

// gfx1250 async copies: global -> LDS, tracked by ASYNCcnt.
__device__ __forceinline__ void async_ld_f32(float* lds_dst, const float* gptr) {
    unsigned la = (unsigned)(unsigned long long)lds_dst; // LDS byte offset
    asm volatile("global_load_async_to_lds_b32 %0, %1, off"
                 :: "v"(la), "v"(gptr) : "memory");
}
__device__ __forceinline__ void async_ld_f128(float* lds_dst, const float* gptr) {
    unsigned la = (unsigned)(unsigned long long)lds_dst;
    asm volatile("global_load_async_to_lds_b128 %0, %1, off"
                 :: "v"(la), "v"(gptr) : "memory");
}
__device__ __forceinline__ void wait_async_all() {
    asm volatile("s_wait_asynccnt 0x0" ::: "memory");
}

// one of {I, J, I*I, J*J, I*J} read from the staged LDS tiles
__device__ __forceinline__ float mapval(const float* sI, const float* sJ,
                                        int map, int idx) {
    if (map == 0) return sI[idx];
    if (map == 1) return sJ[idx];
    if (map == 2) { float v = sI[idx]; return v * v; }
    if (map == 3) { float v = sJ[idx]; return v * v; }
    return sI[idx] * sJ[idx];
}

__global__ __launch_bounds__(NTHREADS) void ncc_kernel(
    const float* __restrict__ ytf, const float* __restrict__ ypf,
    const float* __restrict__ ytb, const float* __restrict__ ypb,
    const unsigned char* __restrict__ self_, const unsigned char* __restrict__ selb_,
    float* __restrict__ acc /* [2*4] per (dir,b) cc sums */)
{
    __shared__ float sI[TH * TW];            // 23040 B
    __shared__ float sJ[TH * TW];            // 23040 B
    __shared__ float sT[TILE_ROWS * TW];     // 18432 B  (total 64512 B)

    const int tid = threadIdx.x;
    const int bx  = blockIdx.x;              // 0..3   column tile
    const int by  = blockIdx.y;              // 0..15  row tile
    const int bz  = blockIdx.z;              // 0..63  (dir, img)
    const int dir = bz >> 5;
    const int img = bz & 31;                 // b*8 + n
    const int b   = img >> 3;

    const float* I = (dir ? ytb : ytf) + (size_t)img * (IMG_H * IMG_W);
    const float* J = (dir ? ypb : ypf) + (size_t)img * (IMG_H * IMG_W);
    const float selv = (float)((dir ? selb_ : self_)[img]);

    const int X0 = bx * TILE_COLS;
    const int Y0 = by * TILE_ROWS;

    // ---- stage I, J tiles (with halo) into LDS via async copies ----
    // 4-float groups; b128 fast path when fully in-bounds & within 136 cols.
    const int NG = TH * (TW / 4);            // 40 * 36 groups
    for (int g = tid; g < NG; g += NTHREADS) {
        int lr  = g / (TW / 4);
        int lc4 = (g - lr * (TW / 4)) * 4;   // 0,4,...,140
        int gr  = Y0 - 4 + lr;
        int gcB = X0 - 4 + lc4;
        int li  = lr * TW + lc4;
        bool rowOk = ((unsigned)gr < (unsigned)IMG_H);
        if (rowOk && lc4 < 136 && gcB >= 0 && gcB + 3 < IMG_W) {
            size_t gi = (size_t)gr * IMG_W + (size_t)gcB;
            async_ld_f128(&sI[li], I + gi);
            async_ld_f128(&sJ[li], J + gi);
        } else {
#pragma unroll
            for (int e = 0; e < 4; ++e) {
                int gc = gcB + e;
                if (rowOk && (lc4 + e) < 136 && (unsigned)gc < (unsigned)IMG_W) {
                    size_t gi = (size_t)gr * IMG_W + (size_t)gc;
                    async_ld_f32(&sI[li + e], I + gi);
                    async_ld_f32(&sJ[li + e], J + gi);
                } else {
                    sI[li + e] = 0.0f;       // SAME zero padding / pad cols
                    sJ[li + e] = 0.0f;
                }
            }
        }
    }
    wait_async_all();
    __syncthreads();

    const int lane = tid & 31;
    const int wave = tid >> 5;               // 0..15
    const int half = lane >> 4;              // lane-half for WMMA fragments
    const int n    = lane & 15;              // N column within fragment
    const int m    = n;                      // A-matrix row index

    const int hr = tid >> 4;                 // 0..31 row for horizontal pass
    const int cb = (tid & 15) * 8;           // col base (owns 8 pixels)

    // Band A-fragments (out row m sums input rows k in [m, m+8]); constant
    // across maps / tiles / row-bands -> hoisted.
    v2f afrag[6];
#pragma unroll
    for (int c = 0; c < 6; ++c) {
        int k0 = 4 * c + 2 * half;
        afrag[c].x = (k0     >= m && k0     <= m + 8) ? 1.0f : 0.0f;
        afrag[c].y = (k0 + 1 >= m && k0 + 1 <= m + 8) ? 1.0f : 0.0f;
    }

    float S[5][8];

#pragma unroll
    for (int map = 0; map < 5; ++map) {
        // ---- vertical 9-tap box sum via WMMA -------------------------------
        // 18 tiles: 2 row-bands x 9 col-tiles over 16 waves (waves 0,1 take 2)
        for (int tt = wave; tt < 18; tt += 16) {
            int rb = (tt >= 9) ? 1 : 0;
            int t  = tt - rb * 9;
            v8f acc8 = {0.f, 0.f, 0.f, 0.f, 0.f, 0.f, 0.f, 0.f};
#pragma unroll
            for (int c = 0; c < 6; ++c) {
                int k0 = 4 * c + 2 * half;
                int base = (16 * rb + k0) * TW + 16 * t + n;
                v2f bb;
                bb.x = mapval(sI, sJ, map, base);
                bb.y = mapval(sI, sJ, map, base + TW);
                acc8 = __builtin_amdgcn_wmma_f32_16x16x4_f32(
                    false, afrag[c], false, bb, (short)0, acc8, false, false);
            }
            // D layout: VGPR r -> M = r + 8*half, N = lane&15
#pragma unroll
            for (int r = 0; r < 8; ++r)
                sT[(16 * rb + r + 8 * half) * TW + 16 * t + n] = acc8[r];
        }
        __syncthreads();

        // ---- horizontal 9-tap running sum from sT (shifted coords) ----
        {
            const float* Trow = &sT[hr * TW + cb];
            float run = 0.0f;
#pragma unroll
            for (int k = 0; k < 9; ++k) run += Trow[k];
            S[map][0] = run;
#pragma unroll
            for (int p = 1; p < 8; ++p) {
                run += Trow[p + 8] - Trow[p - 1];
                S[map][p] = run;
            }
        }
        __syncthreads();
    }

    // ---- per-pixel cc and block partial sum ----
    float local = 0.0f;
#pragma unroll
    for (int p = 0; p < 8; ++p) {
        float Is = S[0][p], Js = S[1][p];
        float I2 = S[2][p], J2 = S[3][p], IJ = S[4][p];
        float uI = Is * (1.0f / WIN_AREA);
        float uJ = Js * (1.0f / WIN_AREA);
        float cross = IJ - uJ * Is - uI * Js + uI * uJ * WIN_AREA;
        float Iv = I2 - 2.0f * uI * Is + uI * uI * WIN_AREA;
        float Jv = J2 - 2.0f * uJ * Js + uJ * uJ * WIN_AREA;
        local += (cross * cross) / (Iv * Jv + 1e-5f);
    }

    // block tree-reduce (reuse sT: 4608 floats >= 512)
    sT[tid] = local;
    __syncthreads();
    for (int s = NTHREADS / 2; s > 0; s >>= 1) {
        if (tid < s) sT[tid] += sT[tid + s];
        __syncthreads();
    }
    if (tid == 0) atomicAdd(&acc[dir * 4 + b], sT[0] * selv);
}

// ---------------- grad loss: sum of squared finite differences --------------
__global__ __launch_bounds__(256) void grad_kernel(
    const float* __restrict__ dff, const float* __restrict__ dfb,
    float* __restrict__ acc /* acc[8+dir] */)
{
    const int dir = blockIdx.y;
    const float* p = dir ? dfb : dff;
    const size_t N = (size_t)4 * 2 * IMG_H * IMG_W;
    float local = 0.0f;
    for (size_t i = (size_t)blockIdx.x * 256 + threadIdx.x; i < N;
         i += (size_t)gridDim.x * 256) {
        int w = (int)(i & (IMG_W - 1));
        int h = (int)((i >> 9) & (IMG_H - 1));
        float v = p[i];
        if (w < IMG_W - 1) { float d = p[i + 1] - v;      local += d * d; }
        if (h < IMG_H - 1) { float d = p[i + IMG_W] - v;  local += d * d; }
    }
    __shared__ float red[256];
    red[threadIdx.x] = local;
    __syncthreads();
    for (int s = 128; s > 0; s >>= 1) {
        if (threadIdx.x < s) red[threadIdx.x] += red[threadIdx.x + s];
        __syncthreads();
    }
    if (threadIdx.x == 0) atomicAdd(&acc[8 + dir], red[0]);
}

// ---------------- init / finalize -------------------------------------------
__global__ void zero_kernel(float* ws) {
    if (threadIdx.x < 16) ws[threadIdx.x] = 0.0f;
}

__global__ void final_kernel(const unsigned char* __restrict__ sf,
                             const unsigned char* __restrict__ sb,
                             const float* __restrict__ acc,
                             float* __restrict__ out)
{
    if (threadIdx.x != 0) return;
    float ncc[2];
    for (int dir = 0; dir < 2; ++dir) {
        const unsigned char* s = dir ? sb : sf;
        float tot = 0.0f;
        for (int b = 0; b < 4; ++b) {
            float ms = 0.0f;
            for (int n = 0; n < 8; ++n) ms += (float)s[b * 8 + n];
            tot += acc[dir * 4 + b] / (ms * (float)(IMG_H * IMG_W));
        }
        ncc[dir] = -0.25f * tot;   // -mean over B
    }
    float ncc_part = 0.5f * (ncc[0] + ncc[1]);

    // grad_loss = ((mean dx^2 + mean dy^2)/2) * INT_DOWNSIZE(2) = sum / cnt
    const float cnt = 4.0f * 2.0f * 512.0f * 511.0f;
    float gl0 = acc[8] / cnt;
    float gl1 = acc[9] / cnt;
    float grad_part = 0.5f * 0.01f * gl0 + 0.5f * 0.01f * gl1;

    out[0] = ncc_part + grad_part;
    out[1] = ncc_part;
    out[2] = grad_part;
}

extern "C" void kernel_launch(void* const* d_in, const int* in_sizes, int n_in,
                              void* d_out, int out_size, void* d_ws, size_t ws_size,
                              hipStream_t stream) {
    const float*         ytf = (const float*)d_in[0];
    const float*         ypf = (const float*)d_in[1];
    const unsigned char* sf  = (const unsigned char*)d_in[2];
    const float*         ytb = (const float*)d_in[3];
    const float*         ypb = (const float*)d_in[4];
    const unsigned char* sb  = (const unsigned char*)d_in[5];
    const float*         dff = (const float*)d_in[6];
    const float*         dfb = (const float*)d_in[7];
    float* ws  = (float*)d_ws;
    float* out = (float*)d_out;

    zero_kernel<<<1, 32, 0, stream>>>(ws);

    dim3 gNcc(IMG_W / TILE_COLS, IMG_H / TILE_ROWS, 64); // (4, 16, 2*32 images)
    ncc_kernel<<<gNcc, NTHREADS, 0, stream>>>(ytf, ypf, ytb, ypb, sf, sb, ws);

    grad_kernel<<<dim3(1024, 2), 256, 0, stream>>>(dff, dfb, ws);

    final_kernel<<<1, 32, 0, stream>>>(sf, sb, ws, out);
}